// Seq2seqAttentionMechanism_13657996001298
// MI455X (gfx1250) — compile-verified
//
#include <hip/hip_runtime.h>
#include <math.h>

typedef float v2f __attribute__((ext_vector_type(2)));
typedef float v8f __attribute__((ext_vector_type(8)));

typedef int v4i_g __attribute__((vector_size(16)));   // matches builtin's pointee type
typedef __attribute__((address_space(1))) v4i_g as1_v4i;
typedef __attribute__((address_space(3))) v4i_g as3_v4i;

#if __has_builtin(__builtin_amdgcn_global_load_async_to_lds_b128)
#define HAVE_ASYNC_LDS 1
#else
#define HAVE_ASYNC_LDS 0
#endif

constexpr int B_      = 32;
constexpr int S_      = 8192;
constexpr int H_      = 256;
constexpr int NWAVES  = 2;                     // waves per block (wave32)
constexpr int TILES   = 4;                     // 16-row s-tiles per wave
constexpr int CHUNK   = NWAVES * TILES * 16;   // 128 s per block
constexpr int NCHUNK  = S_ / CHUNK;            // 64 chunks per batch
constexpr int TSTRIDE = H_ + 4;                // 260-float row pad: conflict-free A-frag reads
constexpr int REC     = 2 + H_;                // per-chunk partial: [max, sumexp, ctx[256]]

// ---------------- Stage A: stream encoder once, WMMA scores + online softmax ----------------
__global__ __launch_bounds__(NWAVES * 32)
void attn_stageA(const float* __restrict__ dec,
                 const float* __restrict__ enc,
                 float* __restrict__ prob,       // d_out[0 : B*S): raw scores this pass
                 float* __restrict__ partials)   // workspace records
{
  __shared__ float ltile[NWAVES][16][TSTRIDE];  // per-wave 16x256 enc tile (padded)
  __shared__ float decLDS[H_];
  __shared__ float ctxbuf[NWAVES][H_];
  __shared__ float mbuf[NWAVES];
  __shared__ float lbuf[NWAVES];

  const int tid  = threadIdx.x;
  const int lane = tid & 31;
  const int wave = tid >> 5;
  const int b    = blockIdx.y;
  const int chunkBase = blockIdx.x * CHUNK;

  const float* decB = dec + b * H_;
  const float* encB = enc + (size_t)b * S_ * H_;

  // decoder vector -> LDS (64 threads x float4 = 256 floats)
  ((float4*)decLDS)[tid] = ((const float4*)decB)[tid];
  __syncthreads();

  const int m16    = lane & 15;     // M row owned by this lane (A / D layout)
  const int rowSel = lane >> 4;     // 0: lanes 0-15, 1: lanes 16-31
  const int klo    = rowSel * 2;    // K offset of this lane's A/B fragment (0 or 2)

  float runMax = -INFINITY;
  float runL   = 0.0f;
  float ctx[8];
#pragma unroll
  for (int i = 0; i < 8; ++i) ctx[i] = 0.0f;

  float (*tile)[TSTRIDE] = ltile[wave];

  for (int it = 0; it < TILES; ++it) {
    const int s0 = chunkBase + (it * NWAVES + wave) * 16;

    // ---- stage 16x256 f32 tile into LDS ----
#if HAVE_ASYNC_LDS
    // CDNA5 async global->LDS: no VGPR round-trip, tracked by ASYNCcnt.
    for (int r2 = 0; r2 < 8; ++r2) {
      const int r = r2 * 2 + rowSel;
      const float* src = encB + (size_t)(s0 + r) * H_;
#pragma unroll
      for (int p = 0; p < 4; ++p) {
        const int c4 = p * 16 + m16;
        __builtin_amdgcn_global_load_async_to_lds_b128(
            (as1_v4i*)(src + c4 * 4),
            (as3_v4i*)&tile[r][c4 * 4],
            /*offset=*/0, /*cpol=*/0);
      }
    }
#if __has_builtin(__builtin_amdgcn_s_wait_asynccnt)
    __builtin_amdgcn_s_wait_asynccnt(0);
#else
    asm volatile("s_wait_asynccnt 0" ::: "memory");
#endif
#else
    // Fallback: coalesced b128 global loads + ds_store_b128
    for (int r2 = 0; r2 < 8; ++r2) {
      const int r = r2 * 2 + rowSel;
      const float4* src = (const float4*)(encB + (size_t)(s0 + r) * H_);
#pragma unroll
      for (int p = 0; p < 4; ++p) {
        const int c4 = p * 16 + m16;
        *(float4*)&tile[r][c4 * 4] = src[c4];
      }
    }
#endif
    // tile is wave-private; DS ops are in-order within a wave -> no barrier needed

    // ---- scores via V_WMMA_F32_16X16X4_F32, accumulating K=256 in chunks of 4 ----
    // A[m][k] = enc[s0+m][off+k] (ISA A-frag layout), B[k][n] = dec[off+k] for all n.
    v8f acc = {0.f, 0.f, 0.f, 0.f, 0.f, 0.f, 0.f, 0.f};
    for (int kc = 0; kc < H_ / 4; ++kc) {
      const int off = kc * 4 + klo;
      v2f a  = *(const v2f*)&tile[m16][off];   // ds_load_b64, all 64 banks hit once
      v2f bb = *(const v2f*)&decLDS[off];      // same-address broadcast per half-wave
      acc = __builtin_amdgcn_wmma_f32_16x16x4_f32(false, a, false, bb,
                                                  (short)0, acc, false, false);
    }

    // Every N column of D equals score[m]. D vgpr j: lanes 0-15 -> M=j, 16-31 -> M=j+8.
    float sc[16];
#pragma unroll
    for (int j = 0; j < 8; ++j) {
      sc[j]     = __shfl(acc[j], 0, 32);
      sc[j + 8] = __shfl(acc[j], 16, 32);
    }

    if (lane == 0) {
#pragma unroll
      for (int j = 0; j < 16; ++j) prob[(size_t)b * S_ + s0 + j] = sc[j];
    }

    // ---- online softmax update + context accumulation out of the LDS tile ----
    float tmax = sc[0];
#pragma unroll
    for (int j = 1; j < 16; ++j) tmax = fmaxf(tmax, sc[j]);
    const float nmax  = fmaxf(runMax, tmax);
    const float scale = __expf(runMax - nmax);   // exp(-inf)=0 handles first tile
    runL *= scale;
#pragma unroll
    for (int i = 0; i < 8; ++i) ctx[i] *= scale;
#pragma unroll 2
    for (int j = 0; j < 16; ++j) {
      const float wj = __expf(sc[j] - nmax);
      runL += wj;
#pragma unroll
      for (int i = 0; i < 8; ++i)
        ctx[i] += wj * tile[j][i * 32 + lane];   // conflict-free: consecutive banks
    }
    runMax = nmax;
  }

  // ---- per-wave partials -> LDS ----
#pragma unroll
  for (int i = 0; i < 8; ++i) ctxbuf[wave][i * 32 + lane] = ctx[i];
  if (lane == 0) { mbuf[wave] = runMax; lbuf[wave] = runL; }
  __syncthreads();

  // ---- merge waves, emit one record per chunk ----
  float M = mbuf[0];
#pragma unroll
  for (int w = 1; w < NWAVES; ++w) M = fmaxf(M, mbuf[w]);

  float* rec = partials + (size_t)(b * NCHUNK + blockIdx.x) * REC;
  for (int h = tid; h < H_; h += NWAVES * 32) {
    float v = 0.f;
#pragma unroll
    for (int w = 0; w < NWAVES; ++w) v += __expf(mbuf[w] - M) * ctxbuf[w][h];
    rec[2 + h] = v;
  }
  if (tid == 0) {
    float L = 0.f;
#pragma unroll
    for (int w = 0; w < NWAVES; ++w) L += __expf(mbuf[w] - M) * lbuf[w];
    rec[0] = M;
    rec[1] = L;
  }
}

// ---------------- Stage B: merge chunk partials, write context, normalize probs ----------------
__global__ __launch_bounds__(256)
void attn_stageB(const float* __restrict__ partials,
                 float* __restrict__ out)   // [0:B*S) probs, [B*S : B*S+B*H) context
{
  const int b   = blockIdx.x;
  const int tid = threadIdx.x;
  const float* recs = partials + (size_t)b * NCHUNK * REC;

  float M = -INFINITY;
  for (int c = 0; c < NCHUNK; ++c) M = fmaxf(M, recs[(size_t)c * REC]);
  float L = 0.f;
  for (int c = 0; c < NCHUNK; ++c)
    L += __expf(recs[(size_t)c * REC] - M) * recs[(size_t)c * REC + 1];

  // context[b][tid] = sum_c exp(m_c - M) * ctx_c[tid]
  float v = 0.f;
  for (int c = 0; c < NCHUNK; ++c)
    v += __expf(recs[(size_t)c * REC] - M) * recs[(size_t)c * REC + 2 + tid];
  out[(size_t)B_ * S_ + (size_t)b * H_ + tid] = v;

  // normalize raw scores in place: p = exp(score - M) / L
  const float inv = 1.0f / L;
  float* probp = out + (size_t)b * S_;
  for (int s = tid; s < S_; s += 256) {
    probp[s] = __expf(probp[s] - M) * inv;
  }
}

extern "C" void kernel_launch(void* const* d_in, const int* in_sizes, int n_in,
                              void* d_out, int out_size, void* d_ws, size_t ws_size,
                              hipStream_t stream) {
  const float* dec = (const float*)d_in[0];   // (32, 256) f32
  const float* enc = (const float*)d_in[1];   // (32, 8192, 256) f32
  float* out = (float*)d_out;                 // 32*8192 probs ++ 32*256 context
  float* ws  = (float*)d_ws;                  // needs 32*64*258*4 ~= 2.1 MB

  dim3 gridA(NCHUNK, B_);
  attn_stageA<<<gridA, NWAVES * 32, 0, stream>>>(dec, enc, out, ws);
  attn_stageB<<<B_, 256, 0, stream>>>(ws, out);
}